// gradmap_cuda_55259049230649
// MI455X (gfx1250) — compile-verified
//
#include <hip/hip_runtime.h>
#include <stdint.h>

// ---------------- constants from the reference ----------------
#define GRD_BOTTOM (1.0f / 1023.0f)
#define GRD_UP     (4.0f / 1023.0f)
#define HALF_C     (0.5f / 1023.0f)
#define INV_GRD_UP (1023.0f / 4.0f)

// ---------------- tile geometry ----------------
#define TW 64          // output tile width
#define TH 64          // output tile height
// x halo: 8 left (16B-aligned b128 loads) covers the needed 5; 8 right (80 total)
// y halo: 5 top / 5 bottom
#define IW (TW + 16)   // 80 input tile width
#define IH (TH + 10)   // 74 input tile height
#define ISW 84         // padded LDS row stride
#define XOFF 8         // input tile x-origin = x0 - XOFF
#define YOFF 5         // input tile y-origin = y0 - YOFF
#define GW (TW + 4)    // 68: gradmap tile incl. halo 2
#define GH (TH + 4)    // 68
#define GSW 72
#define MW (TW + 2)    // 66: maxpool tile incl. halo 1
#define MH (TH + 2)    // 66
#define MSW 68

#define NTHREADS 256

// gfx1250 async global->LDS path (ASYNCcnt); fall back to plain LDS stores
#if defined(__gfx1250__) && \
    __has_builtin(__builtin_amdgcn_global_load_async_to_lds_b32) && \
    __has_builtin(__builtin_amdgcn_s_wait_asynccnt)
#define USE_ASYNC_LDS 1
#else
#define USE_ASYNC_LDS 0
#endif

#if USE_ASYNC_LDS && __has_builtin(__builtin_amdgcn_global_load_async_to_lds_b128)
#define USE_ASYNC_B128 1
#else
#define USE_ASYNC_B128 0
#endif

typedef __attribute__((__vector_size__(4 * sizeof(int)))) int v4i;
typedef __attribute__((address_space(1))) int* gbl_i32_ptr;
typedef __attribute__((address_space(3))) int* lds_i32_ptr;
typedef __attribute__((address_space(1))) v4i* gbl_v4i_ptr;
typedef __attribute__((address_space(3))) v4i* lds_v4i_ptr;

__global__ __launch_bounds__(NTHREADS)
void gradmap_closing_kernel(const float* __restrict__ img,
                            float* __restrict__ out,
                            int H, int W)
{
    __shared__ float s_in[IH * ISW];    // zero-padded input tile
    __shared__ float s_grad[GH * GSW];  // gradmap (-inf outside image)
    __shared__ float s_max[MH * MSW];   // dilation (+inf outside image)

    const int tid = threadIdx.x;
    const int x0 = blockIdx.x * TW;
    const int y0 = blockIdx.y * TH;
    const int64_t plane = blockIdx.z;
    const float* __restrict__ src = img + plane * (int64_t)H * W;
    float* __restrict__ dst = out + plane * (int64_t)H * W;

    // ---- Stage 0: input tile -> LDS (zero padding outside image) ----
    const bool interior = (x0 >= XOFF) && (x0 - XOFF + IW <= W) &&
                          (y0 >= YOFF) && (y0 - YOFF + IH <= H);
    if (interior) {
        // fast path: unpredicated 16B async DMA per lane, no zero-fill needed
        for (int i = tid; i < IH * (IW / 4); i += NTHREADS) {
            const int row = i / (IW / 4);
            const int seg = i - row * (IW / 4);
            const float* gp = src + (int64_t)(y0 + row - YOFF) * W + (x0 - XOFF) + seg * 4;
            float* lp = &s_in[row * ISW + seg * 4];
#if USE_ASYNC_B128
            __builtin_amdgcn_global_load_async_to_lds_b128(
                (gbl_v4i_ptr)(gp), (lds_v4i_ptr)(lp), /*offset=*/0, /*cpol=*/0);
#else
            *(float4*)lp = *(const float4*)gp;
#endif
        }
#if USE_ASYNC_LDS
        __builtin_amdgcn_s_wait_asynccnt(0);
#endif
        __syncthreads();
    } else {
        // border path: zero-fill then predicated per-element copies
        for (int i = tid; i < IH * ISW; i += NTHREADS) s_in[i] = 0.0f;
        __syncthreads();
        for (int i = tid; i < IH * IW; i += NTHREADS) {
            const int ly = i / IW;
            const int lx = i - ly * IW;
            const int gy = y0 + ly - YOFF;
            const int gx = x0 + lx - XOFF;
            if (gy >= 0 && gy < H && gx >= 0 && gx < W) {
                const float* gp = src + (int64_t)gy * W + gx;
                float* lp = &s_in[ly * ISW + lx];
#if USE_ASYNC_LDS
                __builtin_amdgcn_global_load_async_to_lds_b32(
                    (gbl_i32_ptr)(gp), (lds_i32_ptr)(lp), /*offset=*/0, /*cpol=*/0);
#else
                *lp = *gp;
#endif
            }
        }
#if USE_ASYNC_LDS
        __builtin_amdgcn_s_wait_asynccnt(0);
#endif
        __syncthreads();
    }

    // ---- Stage 1: gradmap on tile+2 halo; -inf outside image ----
    for (int i = tid; i < GH * GW; i += NTHREADS) {
        const int ly = i / GW;
        const int lx = i - ly * GW;
        const int iy = y0 + ly - 2;
        const int ix = x0 + lx - 2;
        float v;
        if (iy < 0 || iy >= H || ix < 0 || ix >= W) {
            v = -__builtin_inff();
        } else {
            // s_in origin: (y0-YOFF, x0-XOFF); grad origin: (y0-2, x0-2)
            const float* c = &s_in[(ly + (YOFF - 2)) * ISW + (lx + (XOFF - 2))];
            // x-direction: out[i] = x[i+k] - x[i-k], zero padded
            float g1 = c[1] - c[-1];
            float g2 = c[2] - c[-2];
            float g3 = c[3] - c[-3];
            float m  = (g1 + g2 + g3) * (1.0f / 3.0f);
            float gx = ((g1 < m - HALF_C) || (g1 > m + HALF_C)) ? 0.0f : g1;
            float ax = fabsf(gx);
            ax = ((ax < GRD_BOTTOM) || (ax > GRD_UP)) ? 0.0f : ax;
            // y-direction
            g1 = c[ISW]     - c[-ISW];
            g2 = c[2 * ISW] - c[-2 * ISW];
            g3 = c[3 * ISW] - c[-3 * ISW];
            m  = (g1 + g2 + g3) * (1.0f / 3.0f);
            float gy = ((g1 < m - HALF_C) || (g1 > m + HALF_C)) ? 0.0f : g1;
            float ay = fabsf(gy);
            ay = ((ay < GRD_BOTTOM) || (ay > GRD_UP)) ? 0.0f : ay;
            v = fmaxf(ax, ay) * INV_GRD_UP;
        }
        s_grad[ly * GSW + lx] = v;
    }
    __syncthreads();

    // ---- Stage 2: 3x3 dilation on tile+1 halo; +inf outside image ----
    for (int i = tid; i < MH * MW; i += NTHREADS) {
        const int ly = i / MW;
        const int lx = i - ly * MW;
        const int iy = y0 + ly - 1;
        const int ix = x0 + lx - 1;
        float v;
        if (iy < 0 || iy >= H || ix < 0 || ix >= W) {
            v = __builtin_inff();
        } else {
            const float* c = &s_grad[(ly + 1) * GSW + (lx + 1)];
            float a = fmaxf(fmaxf(c[-GSW - 1], c[-GSW]), c[-GSW + 1]);
            float b = fmaxf(fmaxf(c[-1],       c[0]),    c[1]);
            float d = fmaxf(fmaxf(c[GSW - 1],  c[GSW]),  c[GSW + 1]);
            v = fmaxf(fmaxf(a, b), d);
        }
        s_max[ly * MSW + lx] = v;
    }
    __syncthreads();

    // ---- Stage 3: 3x3 erosion -> output, 4 outputs/thread, b128 stores ----
    for (int i = tid; i < (TH * TW) / 4; i += NTHREADS) {
        const int ly = i / (TW / 4);
        const int lq = i - ly * (TW / 4);
        const int lx = lq * 4;
        const int iy = y0 + ly;
        const int ix = x0 + lx;
        if (iy < H && ix + 3 < W) {
            const float* c = &s_max[(ly + 1) * MSW + (lx + 1)];
            float cm[6];
#pragma unroll
            for (int j = 0; j < 6; ++j)
                cm[j] = fminf(fminf(c[j - 1 - MSW], c[j - 1]), c[j - 1 + MSW]);
            float4 r;
            r.x = fminf(fminf(cm[0], cm[1]), cm[2]);
            r.y = fminf(fminf(cm[1], cm[2]), cm[3]);
            r.z = fminf(fminf(cm[2], cm[3]), cm[4]);
            r.w = fminf(fminf(cm[3], cm[4]), cm[5]);
            *(float4*)(dst + (int64_t)iy * W + ix) = r;
        }
    }
}

extern "C" void kernel_launch(void* const* d_in, const int* in_sizes, int n_in,
                              void* d_out, int out_size, void* d_ws, size_t ws_size,
                              hipStream_t stream) {
    (void)n_in; (void)out_size; (void)d_ws; (void)ws_size;
    const float* img = (const float*)d_in[0];
    float* out = (float*)d_out;

    const int H = 1024;
    const int W = 2048;
    const int planes = in_sizes[0] / (H * W);  // 8 * 3 = 24

    dim3 grid((W + TW - 1) / TW, (H + TH - 1) / TH, planes);  // 32 x 16 x 24
    gradmap_closing_kernel<<<grid, NTHREADS, 0, stream>>>(img, out, H, W);
}